// GraphSageGCN_v3_51342039056724
// MI455X (gfx1250) — compile-verified
//
#include <hip/hip_runtime.h>
#include <cstdint>
#include <cstddef>

// ---------------------------------------------------------------------------
// GraphSage/GCN forward for MI455X (gfx1250, wave32).
//   - Dense GEMMs via V_WMMA_F32_16X16X4_F32 (exact fp32), A tile staged in
//     LDS once per block via GLOBAL_LOAD_ASYNC_TO_LDS_B128. K/NCOLS are
//     template constants so all inner-loop addresses fold into immediate
//     offsets (no per-iteration 64-bit address VALU).
//   - SpMM via edge-parallel gather + global_atomic_add_f32 scatter.
//   - LayerNorm via wave32 shfl_xor reductions.
// ---------------------------------------------------------------------------

static constexpr int N_NODES = 50000;   // multiple of 16 (3125 tiles)
static constexpr int N_EDGES = 800000;
static constexpr int F_IN    = 512;
static constexpr int F_HID   = 256;
static constexpr int F_CAT   = 512;
static constexpr int N_CLS   = 100;
static constexpr float LN_EPS = 1e-5f;

typedef __attribute__((ext_vector_type(2))) float v2f;
typedef __attribute__((ext_vector_type(8))) float v8f;
typedef __attribute__((ext_vector_type(4))) int   v4i;

typedef __attribute__((address_space(1))) v4i* gv4i_p;   // global b128
typedef __attribute__((address_space(3))) v4i* lv4i_p;   // LDS b128

// ---------------------------------------------------------------------------
// C[M,NCOLS] = A[M,K] @ W[K,NCOLS] + bias,  M % 16 == 0, K % 64 == 0.
// Block = 8 waves, one 16-row tile; A tile (16 x K, contiguous) staged in LDS.
// Each wave: two 16-col tiles (A reuse) x two split-K accumulators (ILP).
// Out-of-range columns are CLAMPED (not branched): garbage only reaches
// D columns that are never stored, so the inner loop is branch-free.
// ---------------------------------------------------------------------------
template <int K, int NCOLS>
__global__ __launch_bounds__(256) void gemm_bias_wmma_f32(
    const float* __restrict__ A, const float* __restrict__ W,
    const float* __restrict__ bias, float* __restrict__ C)
{
  __shared__ float As[16 * K];                     // 16 or 32 KB
  const int tid   = threadIdx.x;
  const int mtile = blockIdx.x;

  // ---- async stage of contiguous 16 x K A tile into LDS ------------------
  {
    const float4* Ag  = (const float4*)(A + (size_t)mtile * 16 * K);
    float4*       As4 = (float4*)As;
    constexpr int n4  = (16 * K) >> 2;
#if __has_builtin(__builtin_amdgcn_global_load_async_to_lds_b128)
#pragma unroll
    for (int i = 0; i < n4 / 256; ++i) {
      __builtin_amdgcn_global_load_async_to_lds_b128(
          (gv4i_p)(Ag + i * 256 + tid), (lv4i_p)(As4 + i * 256 + tid), 0, 0);
    }
    asm volatile("s_wait_asynccnt 0" ::: "memory");
#else
#pragma unroll
    for (int i = 0; i < n4 / 256; ++i) As4[i * 256 + tid] = Ag[i * 256 + tid];
#endif
  }
  __syncthreads();

  const int lane = tid & 31;
  const int wave = tid >> 5;
  constexpr int ntiles = (NCOLS + 15) >> 4;
  const int nt0 = wave * 2;
  if (nt0 >= ntiles) return;                       // after barrier; uniform/wave

  const int r     = lane & 15;
  const int khalf = lane >> 4;                     // 0 or 1
  const int c0    = nt0 * 16 + r;
  const int c1    = c0 + 16;
  const bool ok0  = c0 < NCOLS;
  const bool ok1  = c1 < NCOLS;
  const int c0s   = ok0 ? c0 : 0;                  // clamped (always legal)
  const int c1s   = ok1 ? c1 : 0;

  const float* Alds = As + r * K + 2 * khalf;      // LDS, compile-time offsets
  const float* wp0  = W + khalf * NCOLS + c0s;     // global, compile-time offsets
  const float* wp1  = W + khalf * NCOLS + c1s;

  v8f acc0a = {}, acc0b = {}, acc1a = {}, acc1b = {};

#pragma unroll 4
  for (int kk = 0; kk < K; kk += 8) {
    const v2f a0 = *(const v2f*)(Alds + kk);       // K = kk..kk+3 across wave
    const v2f a1 = *(const v2f*)(Alds + kk + 4);   // K = kk+4..kk+7
    v2f b00, b01, b10, b11;
    b00.x = wp0[(kk + 0) * NCOLS];
    b00.y = wp0[(kk + 2) * NCOLS];
    b01.x = wp0[(kk + 4) * NCOLS];
    b01.y = wp0[(kk + 6) * NCOLS];
    b10.x = wp1[(kk + 0) * NCOLS];
    b10.y = wp1[(kk + 2) * NCOLS];
    b11.x = wp1[(kk + 4) * NCOLS];
    b11.y = wp1[(kk + 6) * NCOLS];
    // 4 independent WMMA chains (2 col tiles x 2 split-K accumulators)
    acc0a = __builtin_amdgcn_wmma_f32_16x16x4_f32(false, a0, false, b00,
                                                  (short)0, acc0a, false, false);
    acc1a = __builtin_amdgcn_wmma_f32_16x16x4_f32(false, a0, false, b10,
                                                  (short)0, acc1a, false, false);
    acc0b = __builtin_amdgcn_wmma_f32_16x16x4_f32(false, a1, false, b01,
                                                  (short)0, acc0b, false, false);
    acc1b = __builtin_amdgcn_wmma_f32_16x16x4_f32(false, a1, false, b11,
                                                  (short)0, acc1b, false, false);
  }

  const v8f s0 = acc0a + acc0b;
  const v8f s1 = acc1a + acc1b;
  // D layout: VGPR i holds M = i (lanes 0-15) / M = 8+i (lanes 16-31), N = lane&15
  const int   mbase = mtile * 16 + khalf * 8;
  const float bv0   = bias[c0s];
  const float bv1   = bias[c1s];
  float* Cp0 = C + (size_t)mbase * NCOLS + c0;
  float* Cp1 = C + (size_t)mbase * NCOLS + c1;
#pragma unroll
  for (int i = 0; i < 8; ++i) {
    if (ok0) Cp0[i * NCOLS] = s0[i] + bv0;
    if (ok1) Cp1[i * NCOLS] = s1[i] + bv1;
  }
}

// ---------------------------------------------------------------------------
// SpMM: out[rows[e], :] += vals[e] * feat[cols[e], :]   (F_HID = 256 wide)
// One wave per edge: 32 lanes x 2 float4 = 256 floats. Scatter via hardware
// f32 global atomics (output fits in 192MB L2). Prefetch next gather row.
// ---------------------------------------------------------------------------
__global__ __launch_bounds__(256) void spmm_edge_atomic(
    const int* __restrict__ rows, const int* __restrict__ cols,
    const float* __restrict__ vals, const float* __restrict__ feat,
    float* __restrict__ out, int E)
{
  const int lane   = threadIdx.x & 31;
  const int warp   = blockIdx.x * (blockDim.x >> 5) + (threadIdx.x >> 5);
  const int nwarps = gridDim.x * (blockDim.x >> 5);
  for (int e = warp; e < E; e += nwarps) {
    const int   dst = rows[e];
    const int   src = cols[e];
    const float v   = vals[e];
    if (e + nwarps < E) {
      // global_prefetch_b8 of next edge's feature row (one line per lane)
      __builtin_prefetch(feat + (size_t)cols[e + nwarps] * F_HID + lane * 8, 0, 0);
    }
    const float4* fp = (const float4*)(feat + (size_t)src * F_HID);
    float*        op = out + (size_t)dst * F_HID;
#pragma unroll
    for (int j = 0; j < 2; ++j) {
      const float4 x    = fp[lane + 32 * j];
      const int    base = (lane + 32 * j) * 4;
      unsafeAtomicAdd(op + base + 0, v * x.x);
      unsafeAtomicAdd(op + base + 1, v * x.y);
      unsafeAtomicAdd(op + base + 2, v * x.z);
      unsafeAtomicAdd(op + base + 3, v * x.w);
    }
  }
}

// ---------------------------------------------------------------------------
// outcat[row, 0:256]=feat part, [256:512]=agg part; LayerNorm over all 512
// then ReLU. One wave32 per row; 16 elements/lane; shfl_xor tree reduction.
// ---------------------------------------------------------------------------
__global__ __launch_bounds__(256) void ln_relu_cat_kernel(
    const float* __restrict__ feat, const float* __restrict__ agg,
    const float* __restrict__ g, const float* __restrict__ b,
    float* __restrict__ outcat, int Nrows)
{
  const int lane = threadIdx.x & 31;
  const int row  = blockIdx.x * (blockDim.x >> 5) + (threadIdx.x >> 5);
  if (row >= Nrows) return;
  const float4* fp = (const float4*)(feat + (size_t)row * F_HID);
  const float4* ap = (const float4*)(agg  + (size_t)row * F_HID);
  float4 x[4];
  x[0] = fp[2 * lane];  x[1] = fp[2 * lane + 1];
  x[2] = ap[2 * lane];  x[3] = ap[2 * lane + 1];
  float s = 0.f, s2 = 0.f;
#pragma unroll
  for (int j = 0; j < 4; ++j) {
    s  += x[j].x + x[j].y + x[j].z + x[j].w;
    s2 += x[j].x * x[j].x + x[j].y * x[j].y + x[j].z * x[j].z + x[j].w * x[j].w;
  }
#pragma unroll
  for (int off = 16; off >= 1; off >>= 1) {
    s  += __shfl_xor(s,  off, 32);
    s2 += __shfl_xor(s2, off, 32);
  }
  const float mu  = s  * (1.f / 512.f);
  const float var = s2 * (1.f / 512.f) - mu * mu;   // population variance
  const float rs  = rsqrtf(var + LN_EPS);
  float* o = outcat + (size_t)row * F_CAT;
#pragma unroll
  for (int j = 0; j < 4; ++j) {
    const int idx = (j < 2) ? (2 * lane + j) * 4
                            : 256 + (2 * lane + (j - 2)) * 4;
    const float vv[4] = {x[j].x, x[j].y, x[j].z, x[j].w};
#pragma unroll
    for (int k = 0; k < 4; ++k) {
      const float yv = (vv[k] - mu) * rs * g[idx + k] + b[idx + k];
      o[idx + k] = yv > 0.f ? yv : 0.f;
    }
  }
}

__global__ __launch_bounds__(256) void relu4_kernel(
    const float4* __restrict__ in, float4* __restrict__ out, int n4)
{
  const int i = blockIdx.x * blockDim.x + threadIdx.x;
  if (i < n4) {
    float4 x = in[i];
    x.x = fmaxf(x.x, 0.f);
    x.y = fmaxf(x.y, 0.f);
    x.z = fmaxf(x.z, 0.f);
    x.w = fmaxf(x.w, 0.f);
    out[i] = x;
  }
}

// ---------------------------------------------------------------------------
extern "C" void kernel_launch(void* const* d_in, const int* in_sizes, int n_in,
                              void* d_out, int out_size, void* d_ws, size_t ws_size,
                              hipStream_t stream)
{
  (void)in_sizes; (void)n_in; (void)out_size; (void)ws_size;

  const float* x    = (const float*)d_in[0];
  const int*   adjr = (const int*)  d_in[1];   // [4,E]
  const int*   adjc = (const int*)  d_in[2];   // [4,E]
  const float* adjv = (const float*)d_in[3];   // [4,E]
  const float* W0   = (const float*)d_in[4];
  const float* b0   = (const float*)d_in[5];
  const float* ln0g = (const float*)d_in[6];
  const float* ln0b = (const float*)d_in[7];
  const float* W1   = (const float*)d_in[8];
  const float* b1   = (const float*)d_in[9];
  const float* W2   = (const float*)d_in[10];
  const float* b2   = (const float*)d_in[11];
  const float* ln2g = (const float*)d_in[12];
  const float* ln2b = (const float*)d_in[13];
  const float* W3   = (const float*)d_in[14];
  const float* b3   = (const float*)d_in[15];
  const float* Wout = (const float*)d_in[16];
  const float* bout = (const float*)d_in[17];
  float* out = (float*)d_out;

  // workspace layout
  float* B_t   = (float*)d_ws;                          // [N,256] GEMM out / feat
  float* B_agg = B_t   + (size_t)N_NODES * F_HID;       // [N,256] SpMM accumulator
  float* B_h   = B_agg + (size_t)N_NODES * F_HID;       // [N,256] post-ReLU h
  float* B_cat = B_h   + (size_t)N_NODES * F_HID;       // [N,512] concat / LN out

  const dim3 blk(256);
  const dim3 g_gemm(N_NODES / 16, 1);
  const int    spmm_blocks = 2048;
  const int    ln_blocks   = (N_NODES + 7) / 8;
  const int    n4          = N_NODES * F_HID / 4;
  const int    relu_blocks = (n4 + 255) / 256;
  const size_t aggBytes    = (size_t)N_NODES * F_HID * sizeof(float);

  // ---- layer 0: GraphSage(512 -> 256) + LN(512) + ReLU -> B_cat
  gemm_bias_wmma_f32<F_IN, F_HID><<<g_gemm, blk, 0, stream>>>(x, W0, b0, B_t);
  (void)hipMemsetAsync(B_agg, 0, aggBytes, stream);
  spmm_edge_atomic<<<spmm_blocks, blk, 0, stream>>>(
      adjr + 0 * (size_t)N_EDGES, adjc + 0 * (size_t)N_EDGES,
      adjv + 0 * (size_t)N_EDGES, B_t, B_agg, N_EDGES);
  ln_relu_cat_kernel<<<ln_blocks, blk, 0, stream>>>(B_t, B_agg, ln0g, ln0b, B_cat, N_NODES);

  // ---- layer 1: GCN(512 -> 256) + ReLU -> B_h
  gemm_bias_wmma_f32<F_CAT, F_HID><<<g_gemm, blk, 0, stream>>>(B_cat, W1, b1, B_t);
  (void)hipMemsetAsync(B_agg, 0, aggBytes, stream);
  spmm_edge_atomic<<<spmm_blocks, blk, 0, stream>>>(
      adjr + 1 * (size_t)N_EDGES, adjc + 1 * (size_t)N_EDGES,
      adjv + 1 * (size_t)N_EDGES, B_t, B_agg, N_EDGES);
  relu4_kernel<<<relu_blocks, blk, 0, stream>>>((const float4*)B_agg, (float4*)B_h, n4);

  // ---- layer 2: GraphSage(256 -> 256) + LN(512) + ReLU -> B_cat
  gemm_bias_wmma_f32<F_HID, F_HID><<<g_gemm, blk, 0, stream>>>(B_h, W2, b2, B_t);
  (void)hipMemsetAsync(B_agg, 0, aggBytes, stream);
  spmm_edge_atomic<<<spmm_blocks, blk, 0, stream>>>(
      adjr + 2 * (size_t)N_EDGES, adjc + 2 * (size_t)N_EDGES,
      adjv + 2 * (size_t)N_EDGES, B_t, B_agg, N_EDGES);
  ln_relu_cat_kernel<<<ln_blocks, blk, 0, stream>>>(B_t, B_agg, ln2g, ln2b, B_cat, N_NODES);

  // ---- layer 3: GCN(512 -> 256) + ReLU -> B_h
  gemm_bias_wmma_f32<F_CAT, F_HID><<<g_gemm, blk, 0, stream>>>(B_cat, W3, b3, B_t);
  (void)hipMemsetAsync(B_agg, 0, aggBytes, stream);
  spmm_edge_atomic<<<spmm_blocks, blk, 0, stream>>>(
      adjr + 3 * (size_t)N_EDGES, adjc + 3 * (size_t)N_EDGES,
      adjv + 3 * (size_t)N_EDGES, B_t, B_agg, N_EDGES);
  relu4_kernel<<<relu_blocks, blk, 0, stream>>>((const float4*)B_agg, (float4*)B_h, n4);

  // ---- output head: h @ Wout + bout (NCOLS = 100, clamped tiles)
  gemm_bias_wmma_f32<F_HID, N_CLS><<<g_gemm, blk, 0, stream>>>(B_h, Wout, bout, out);
}